// TreeModel_72456098283564
// MI455X (gfx1250) — compile-verified
//
#include <hip/hip_runtime.h>
#include <math.h>

// Problem constants (match reference)
#define B_   32
#define D_   12
#define H_   256
#define X_   256
#define NCLS 20
#define M_   ((1 << D_) - 1)          // 4095
#define N_   (B_ * M_)                // 131040
#define MAXL (B_ * (1 << (D_ - 1)))   // 65536 rows at leaf level

typedef __attribute__((ext_vector_type(2))) float v2f;
typedef __attribute__((ext_vector_type(8))) float v8f;

__device__ __forceinline__ float sigmoidf_(float x) {
    return 1.0f / (1.0f + __expf(-x));
}

__device__ __forceinline__ v8f wmma4(v2f a, v2f b, v8f c) {
    return __builtin_amdgcn_wmma_f32_16x16x4_f32(false, a, false, b,
                                                 (short)0, c, false, false);
}

// ---------------------------------------------------------------------------
// GEMM1: iou[r, 0:768] = A(r) @ W^T + b_iou
//   leaf level (l == D-1): A = masked embedding gather, W = W_iou
//   internal levels:       A = h_sum (child-sum),       W = U_iou
// (x_mask is 1 only on leaves -> the other term is exactly zero.)
//
// Wave tile = 32x32 (2x2 of 16x16 WMMA). K walked in 8-wide blocks with a
// permuted K order so each lane's two WMMA steps read ONE contiguous b128:
//   lanes 0-15 own k..k+3, lanes 16-31 own k+4..k+7 of each 8-block.
// grid = (3, Bl/32), block = 256 (8 waves cover 256 cols; 3 blocks cover 768).
// ---------------------------------------------------------------------------
__global__ void __launch_bounds__(256)
gemm_iou_kernel(const int* __restrict__ x, const int* __restrict__ xmask,
                const float* __restrict__ Emb, const float* __restrict__ W_iou,
                const float* __restrict__ b_iou, const float* __restrict__ U_iou,
                const float* __restrict__ h_sum, float* __restrict__ iou,
                int l, int leaf) {
    const int lane  = threadIdx.x & 31;
    const int wv    = threadIdx.x >> 5;              // 0..7
    const int col0  = (blockIdx.x * 8 + wv) * 32;    // 0..736
    const int row0  = blockIdx.y * 32;
    const int mrow  = lane & 15;
    const int khalf = lane >> 4;                     // K sub-block select
    const int ncol  = lane & 15;

    const int r0 = row0 + mrow;                      // A row, tile 0
    const int r1 = r0 + 16;                          // A row, tile 1
    const float* __restrict__ A0;
    const float* __restrict__ A1;
    const float* __restrict__ W;
    float fm0 = 1.0f, fm1 = 1.0f;

    if (leaf) {
        const int t0 = r0 >> l, t1 = r1 >> l, msk = (1 << l) - 1, base = msk;
        const int n0 = t0 * M_ + base + (r0 & msk);
        const int n1 = t1 * M_ + base + (r1 & msk);
        const int m0 = xmask[n0], m1 = xmask[n1];
        fm0 = (float)m0;  fm1 = (float)m1;
        A0 = Emb + (size_t)(x[n0] * m0) * X_;
        A1 = Emb + (size_t)(x[n1] * m1) * X_;
        W  = W_iou;
    } else {
        A0 = h_sum + (size_t)r0 * H_;
        A1 = h_sum + (size_t)r1 * H_;
        W  = U_iou;
    }
    const float* __restrict__ B0 = W + (size_t)(col0 + ncol) * H_;
    const float* __restrict__ B1 = W + (size_t)(col0 + 16 + ncol) * H_;

    __builtin_prefetch(A0, 0, 3);
    __builtin_prefetch(A1, 0, 3);

    v8f acc00 = {}, acc01 = {}, acc10 = {}, acc11 = {};

    #pragma unroll 2
    for (int k = 0; k < H_; k += 8) {
        const int q = (k >> 2) + khalf;              // float4 index, 16B aligned
        float4 a0 = ((const float4*)A0)[q];
        float4 a1 = ((const float4*)A1)[q];
        const float4 b0 = ((const float4*)B0)[q];
        const float4 b1 = ((const float4*)B1)[q];
        if (leaf) {
            a0.x *= fm0; a0.y *= fm0; a0.z *= fm0; a0.w *= fm0;
            a1.x *= fm1; a1.y *= fm1; a1.z *= fm1; a1.w *= fm1;
        }
        const v2f a0l = {a0.x, a0.y}, a0h = {a0.z, a0.w};
        const v2f a1l = {a1.x, a1.y}, a1h = {a1.z, a1.w};
        const v2f b0l = {b0.x, b0.y}, b0h = {b0.z, b0.w};
        const v2f b1l = {b1.x, b1.y}, b1h = {b1.z, b1.w};
        acc00 = wmma4(a0l, b0l, acc00);
        acc01 = wmma4(a0l, b1l, acc01);
        acc10 = wmma4(a1l, b0l, acc10);
        acc11 = wmma4(a1l, b1l, acc11);
        acc00 = wmma4(a0h, b0h, acc00);
        acc01 = wmma4(a0h, b1h, acc01);
        acc10 = wmma4(a1h, b0h, acc10);
        acc11 = wmma4(a1h, b1h, acc11);
    }

    // C/D layout: vgpr v -> (m = v + 8*(lane>=16), n = lane&15)
    const float bias0 = b_iou[col0 + ncol];
    const float bias1 = b_iou[col0 + 16 + ncol];
    const int mHi = (lane >> 4) * 8;
    #pragma unroll
    for (int v = 0; v < 8; ++v) {
        const int m = mHi + v;
        float* row = iou + (size_t)(row0 + m) * (3 * H_);
        row[col0 + ncol]      = acc00[v] + bias0;
        row[col0 + 16 + ncol] = acc01[v] + bias1;
        row += (size_t)16 * (3 * H_);
        row[col0 + ncol]      = acc10[v] + bias0;
        row[col0 + 16 + ncol] = acc11[v] + bias1;
    }
}

// ---------------------------------------------------------------------------
// Pointwise gates: c = sig(i)*tanh(u) + fc_sum ; h = sig(o)*tanh(c)
// ---------------------------------------------------------------------------
__global__ void __launch_bounds__(256)
gates_kernel(const float* __restrict__ iou, const float* __restrict__ fc_sum,
             float* __restrict__ c_lvl, float* __restrict__ h_all,
             int l, int hasFC) {
    const int r = blockIdx.x;
    const int h = threadIdx.x;
    const size_t base = (size_t)r * (3 * H_);
    const float i = iou[base + h];
    const float o = iou[base + H_ + h];
    const float u = iou[base + 2 * H_ + h];
    float c = sigmoidf_(i) * tanhf(u);
    if (hasFC) c += fc_sum[(size_t)r * H_ + h];
    const float hv = sigmoidf_(o) * tanhf(c);
    c_lvl[(size_t)r * H_ + h] = c;
    const int node = (r >> l) * M_ + ((1 << l) - 1) + (r & ((1 << l) - 1));
    h_all[(size_t)node * H_ + h] = hv;
}

// ---------------------------------------------------------------------------
// GEMM2: fc[r, 0:256] = sigmoid( h[r] @ U_f^T + b_f ) * c[r]
// Wave tile = 32x32; 8 waves cover all 256 cols. grid = (1, Bl/32).
// ---------------------------------------------------------------------------
__global__ void __launch_bounds__(256)
gemm_f_kernel(const float* __restrict__ h_all, const float* __restrict__ U_f,
              const float* __restrict__ b_f, const float* __restrict__ c_lvl,
              float* __restrict__ fc_lvl, int l) {
    const int lane  = threadIdx.x & 31;
    const int wv    = threadIdx.x >> 5;
    const int col0  = wv * 32;                       // 0..224
    const int row0  = blockIdx.y * 32;
    const int mrow  = lane & 15;
    const int khalf = lane >> 4;
    const int ncol  = lane & 15;

    const int r0 = row0 + mrow, r1 = r0 + 16;
    const int msk = (1 << l) - 1;
    const int n0 = (r0 >> l) * M_ + msk + (r0 & msk);
    const int n1 = (r1 >> l) * M_ + msk + (r1 & msk);
    const float* __restrict__ A0 = h_all + (size_t)n0 * H_;
    const float* __restrict__ A1 = h_all + (size_t)n1 * H_;
    const float* __restrict__ B0 = U_f + (size_t)(col0 + ncol) * H_;
    const float* __restrict__ B1 = U_f + (size_t)(col0 + 16 + ncol) * H_;

    __builtin_prefetch(A0, 0, 3);
    __builtin_prefetch(A1, 0, 3);

    v8f acc00 = {}, acc01 = {}, acc10 = {}, acc11 = {};

    #pragma unroll 2
    for (int k = 0; k < H_; k += 8) {
        const int q = (k >> 2) + khalf;
        const float4 a0 = ((const float4*)A0)[q];
        const float4 a1 = ((const float4*)A1)[q];
        const float4 b0 = ((const float4*)B0)[q];
        const float4 b1 = ((const float4*)B1)[q];
        const v2f a0l = {a0.x, a0.y}, a0h = {a0.z, a0.w};
        const v2f a1l = {a1.x, a1.y}, a1h = {a1.z, a1.w};
        const v2f b0l = {b0.x, b0.y}, b0h = {b0.z, b0.w};
        const v2f b1l = {b1.x, b1.y}, b1h = {b1.z, b1.w};
        acc00 = wmma4(a0l, b0l, acc00);
        acc01 = wmma4(a0l, b1l, acc01);
        acc10 = wmma4(a1l, b0l, acc10);
        acc11 = wmma4(a1l, b1l, acc11);
        acc00 = wmma4(a0h, b0h, acc00);
        acc01 = wmma4(a0h, b1h, acc01);
        acc10 = wmma4(a1h, b0h, acc10);
        acc11 = wmma4(a1h, b1h, acc11);
    }

    const float bias0 = b_f[col0 + ncol];
    const float bias1 = b_f[col0 + 16 + ncol];
    const int mHi = (lane >> 4) * 8;
    #pragma unroll
    for (int v = 0; v < 8; ++v) {
        const int m = mHi + v;
        size_t i0 = (size_t)(row0 + m) * H_;
        fc_lvl[i0 + col0 + ncol]      = sigmoidf_(acc00[v] + bias0) * c_lvl[i0 + col0 + ncol];
        fc_lvl[i0 + col0 + 16 + ncol] = sigmoidf_(acc01[v] + bias1) * c_lvl[i0 + col0 + 16 + ncol];
        i0 += (size_t)16 * H_;
        fc_lvl[i0 + col0 + ncol]      = sigmoidf_(acc10[v] + bias0) * c_lvl[i0 + col0 + ncol];
        fc_lvl[i0 + col0 + 16 + ncol] = sigmoidf_(acc11[v] + bias1) * c_lvl[i0 + col0 + 16 + ncol];
    }
}

// ---------------------------------------------------------------------------
// Parent gather-reduction: each parent row sums its two children (no atomics)
// ---------------------------------------------------------------------------
__global__ void __launch_bounds__(256)
reduce_kernel(const float* __restrict__ h_all, const float* __restrict__ fc_lvl,
              float* __restrict__ h_sum_nxt, float* __restrict__ fc_sum_nxt,
              int l /* child level */) {
    const int pr = blockIdx.x;
    const int h  = threadIdx.x;
    const int lp = l - 1;
    const int b  = pr >> lp;
    const int j  = pr & ((1 << lp) - 1);
    const int cr0   = (b << l) + 2 * j;                 // child row (level l)
    const int node0 = b * M_ + ((1 << l) - 1) + 2 * j;  // child node (global)
    h_sum_nxt[(size_t)pr * H_ + h] =
        h_all[(size_t)node0 * H_ + h] + h_all[(size_t)(node0 + 1) * H_ + h];
    fc_sum_nxt[(size_t)pr * H_ + h] =
        fc_lvl[(size_t)cr0 * H_ + h] + fc_lvl[(size_t)(cr0 + 1) * H_ + h];
}

// ---------------------------------------------------------------------------
// Classifier: out[n, c] = h_all[n] . W_out[c] + b_out[c]   (NCLS=20)
// ---------------------------------------------------------------------------
__global__ void __launch_bounds__(256)
classify_kernel(const float* __restrict__ h_all, const float* __restrict__ W_out,
                const float* __restrict__ b_out, float* __restrict__ out) {
    const int idx = blockIdx.x * blockDim.x + threadIdx.x;
    if (idx >= N_ * NCLS) return;
    const int row = idx / NCLS;
    const int cls = idx % NCLS;
    const float4* __restrict__ hr = (const float4*)(h_all + (size_t)row * H_);
    const float4* __restrict__ wr = (const float4*)(W_out + (size_t)cls * H_);
    float acc = 0.0f;
    #pragma unroll 8
    for (int k = 0; k < H_ / 4; ++k) {
        const float4 a = hr[k];
        const float4 w = wr[k];
        acc = fmaf(a.x, w.x, acc);
        acc = fmaf(a.y, w.y, acc);
        acc = fmaf(a.z, w.z, acc);
        acc = fmaf(a.w, w.w, acc);
    }
    out[idx] = acc + b_out[cls];
}

// ---------------------------------------------------------------------------
extern "C" void kernel_launch(void* const* d_in, const int* in_sizes, int n_in,
                              void* d_out, int out_size, void* d_ws, size_t ws_size,
                              hipStream_t stream) {
    (void)in_sizes; (void)n_in; (void)out_size; (void)ws_size;
    const int*   x      = (const int*)  d_in[0];
    const int*   xmask  = (const int*)  d_in[1];
    const float* Emb    = (const float*)d_in[2];
    const float* W_iou  = (const float*)d_in[3];
    const float* b_iou  = (const float*)d_in[4];
    const float* U_iou  = (const float*)d_in[5];
    const float* U_f    = (const float*)d_in[6];
    const float* b_f    = (const float*)d_in[7];
    const float* W_out  = (const float*)d_in[8];
    const float* b_out  = (const float*)d_in[9];
    float* out = (float*)d_out;

    // Workspace carving (floats). Total ~604 MB.
    float* p      = (float*)d_ws;
    float* h_all  = p;  p += (size_t)N_ * H_;          // 134.2 MB
    float* iou    = p;  p += (size_t)MAXL * 3 * H_;    // 201.3 MB (leaf level)
    float* c_lvl  = p;  p += (size_t)MAXL * H_;        //  67.1 MB
    float* fc_lvl = p;  p += (size_t)MAXL * H_;        //  67.1 MB
    float* hs0    = p;  p += (size_t)(MAXL / 2) * H_;  //  33.6 MB (ping)
    float* hs1    = p;  p += (size_t)(MAXL / 2) * H_;  //  33.6 MB (pong)
    float* fs0    = p;  p += (size_t)(MAXL / 2) * H_;
    float* fs1    = p;  p += (size_t)(MAXL / 2) * H_;

    float* hs_cur = hs0; float* hs_nxt = hs1;
    float* fs_cur = fs0; float* fs_nxt = fs1;

    // Topological sweep: leaves (l = D-1) -> roots (l = 0)
    for (int l = D_ - 1; l >= 0; --l) {
        const int Bl   = B_ << l;        // rows at this level
        const int leaf = (l == D_ - 1);  // leaf: emb pass only; else: h_sum pass

        dim3 g1(3, Bl / 32);             // 24 col tiles of 32 / 8 waves
        gemm_iou_kernel<<<g1, 256, 0, stream>>>(x, xmask, Emb, W_iou, b_iou,
                                                U_iou, hs_cur, iou, l, leaf);

        gates_kernel<<<Bl, 256, 0, stream>>>(iou, fs_cur, c_lvl, h_all, l, !leaf);

        if (l > 0) {
            dim3 g2(1, Bl / 32);         // 8 col tiles of 32 == 8 waves
            gemm_f_kernel<<<g2, 256, 0, stream>>>(h_all, U_f, b_f, c_lvl,
                                                  fc_lvl, l);
            reduce_kernel<<<Bl / 2, 256, 0, stream>>>(h_all, fc_lvl,
                                                      hs_nxt, fs_nxt, l);
            // ping-pong level sums
            float* t;
            t = hs_cur; hs_cur = hs_nxt; hs_nxt = t;
            t = fs_cur; fs_cur = fs_nxt; fs_nxt = t;
        }
    }

    const int total = N_ * NCLS;
    classify_kernel<<<(total + 255) / 256, 256, 0, stream>>>(h_all, W_out,
                                                             b_out, out);
}